// RowMaskHandler_16612933501321
// MI455X (gfx1250) — compile-verified
//
#include <hip/hip_runtime.h>

// ---------------------------------------------------------------------------
// RowMaskHandler for MI455X (gfx1250, wave32).
// B=32, R=4096, C=1024, D=1024. Pure HBM-streaming problem (~1.5 GB traffic,
// ~66 us roofline @ 23.3 TB/s). Row norms^2 computed as diag(A*A^T) with
// V_WMMA_F32_16X16X4_F32: the 16x4 f32 A layout equals the 4x16 B layout of
// its own transpose per-lane, so A and B share the same VGPR pair.
// ---------------------------------------------------------------------------

typedef __attribute__((ext_vector_type(2))) float v2f;
typedef __attribute__((ext_vector_type(4))) float v4f;
typedef __attribute__((ext_vector_type(8))) float v8f;

#define BATCH 32
#define ROWS  4096
#define COLS  1024
#define DLOG  1024

// --------------------------- 1) keep-count ---------------------------------
__global__ void __launch_bounds__(256)
score_kernel(const float* __restrict__ logits, const float* __restrict__ w_score,
             const float* __restrict__ b_score, int* __restrict__ kkeep) {
    const int b = blockIdx.x;
    __shared__ float red[256];
    float s = 0.0f;
    for (int i = threadIdx.x; i < DLOG; i += 256)
        s += logits[b * DLOG + i] * w_score[i];
    red[threadIdx.x] = s;
    __syncthreads();
    for (int off = 128; off > 0; off >>= 1) {
        if (threadIdx.x < off) red[threadIdx.x] += red[threadIdx.x + off];
        __syncthreads();
    }
    if (threadIdx.x == 0) {
        float f = 1.0f / (1.0f + __expf(-(red[0] + b_score[0])));
        int k = (int)(f * (float)ROWS);       // trunc toward zero, like .long()
        if (k < 1) k = 1;
        if (k > ROWS) k = ROWS;
        kkeep[b] = k;
    }
}

// ------------------- 2) row sum-of-squares via WMMA ------------------------
// One wave handles 16 consecutive rows. Per K-chunk of 4: each lane loads a
// float2 laid out exactly as the 16x4 f32 A operand; passing the same regs as
// A and B yields D = W_tile * W_tile^T, whose diagonal accumulates the row
// sums of squares across the 256 chained WMMAs (K = 1024).
__global__ void __launch_bounds__(256)
magsq_kernel(const float* __restrict__ W, float* __restrict__ magsq) {
    const int lane = threadIdx.x & 31;
    const long long waveId = (long long)blockIdx.x * 8 + (threadIdx.x >> 5);
    const long long row0 = waveId * 16;                 // 16 rows per wave
    const int m    = lane & 15;                         // row within tile
    const int ksel = (lane >> 4) << 1;                  // 0 (lanes 0-15) or 2
    const float* p = W + (row0 + m) * (long long)COLS + ksel;

    v8f c = {};                                         // 16x16 f32 accumulator
#pragma unroll 8
    for (int k0 = 0; k0 < COLS; k0 += 4) {
        v2f a = __builtin_nontemporal_load((const v2f*)(p + k0));
        // D = A * A^T + C  (same registers for A and B)
        c = __builtin_amdgcn_wmma_f32_16x16x4_f32(false, a, false, a,
                                                  (short)0, c, false, false);
    }

    // Diagonal (M=N): lanes 0-7 hold diag m=lane in acc[lane];
    // lanes 24-31 hold diag m=lane-16 in acc[lane-24].
    int idx = (lane < 8) ? lane : (lane - 24);
    float d = c[0];
#pragma unroll
    for (int i = 1; i < 8; ++i) d = (idx == i) ? c[i] : d;

    if (lane < 8)        magsq[row0 + lane]       = d;  // squared norm (sqrt
    else if (lane >= 24) magsq[row0 + lane - 16]  = d;  // is monotone: skip it)
}

// ---------------- 3) per-batch k-th largest via bitonic sort ---------------
__global__ void __launch_bounds__(1024)
threshold_kernel(const float* __restrict__ magsq, const int* __restrict__ kkeep,
                 float* __restrict__ thresh) {
    const int b = blockIdx.x;
    __shared__ float s[ROWS];                            // 16 KB of 320 KB LDS
    for (int i = threadIdx.x; i < ROWS; i += 1024)
        s[i] = magsq[b * ROWS + i];
    __syncthreads();
    for (int size = 2; size <= ROWS; size <<= 1) {
        for (int stride = size >> 1; stride > 0; stride >>= 1) {
            for (int t = threadIdx.x; t < ROWS / 2; t += 1024) {
                int i = t + (t & ~(stride - 1));
                int j = i + stride;
                bool up = ((i & size) == 0);             // ascending segment
                float a = s[i], c = s[j];
                bool sw = up ? (a > c) : (a < c);
                if (sw) { s[i] = c; s[j] = a; }
            }
            __syncthreads();
        }
    }
    // descending[k-1] == ascending[ROWS - k]
    if (threadIdx.x == 0) thresh[b] = s[ROWS - kkeep[b]];
}

// --------------------- 4) streaming mask-multiply --------------------------
__global__ void __launch_bounds__(256)
mask_kernel(const float* __restrict__ W, const float* __restrict__ magsq,
            const float* __restrict__ thresh, float* __restrict__ out) {
    const long long nvec = (long long)BATCH * ROWS * (COLS / 4); // float4 count
    long long i      = (long long)blockIdx.x * blockDim.x + threadIdx.x;
    long long stride = (long long)gridDim.x * blockDim.x;
    const v4f* Wv = (const v4f*)W;
    v4f*       Ov = (v4f*)out;
    for (; i < nvec; i += stride) {
        long long row = i >> 8;                 // 256 float4 per row
        int b = (int)(row >> 12);               // 4096 rows per batch
        float msel = (magsq[row] >= thresh[b]) ? 1.0f : 0.0f; // L2-resident
        if (i + stride < nvec)
            __builtin_prefetch(Wv + i + stride, 0, 0);        // global_prefetch_b8
        v4f w = __builtin_nontemporal_load(Wv + i);           // TH=NT b128
        __builtin_nontemporal_store(w * msel, Ov + i);        // TH=NT b128
    }
}

// ---------------------------------------------------------------------------
extern "C" void kernel_launch(void* const* d_in, const int* in_sizes, int n_in,
                              void* d_out, int out_size, void* d_ws, size_t ws_size,
                              hipStream_t stream) {
    const float* W       = (const float*)d_in[0];   // [32,4096,1024]
    const float* logits  = (const float*)d_in[1];   // [32,1024]
    const float* w_score = (const float*)d_in[2];   // [1,1024]
    const float* b_score = (const float*)d_in[3];   // [1]
    float* out = (float*)d_out;

    // Workspace layout: 131072 f32 magsq | 32 f32 thresh | 32 i32 kkeep (~513 KB)
    float* magsq  = (float*)d_ws;
    float* thresh = magsq + BATCH * ROWS;
    int*   kkeep  = (int*)(thresh + BATCH);

    score_kernel<<<BATCH, 256, 0, stream>>>(logits, w_score, b_score, kkeep);

    // 131072 rows / 16 rows-per-wave / 8 waves-per-block = 1024 blocks
    magsq_kernel<<<(BATCH * ROWS) / (16 * 8), 256, 0, stream>>>(W, magsq);

    threshold_kernel<<<BATCH, 1024, 0, stream>>>(magsq, kkeep, thresh);

    // 33.55M float4; 32768 blocks x 256 threads -> 4 float4 per thread
    mask_kernel<<<32768, 256, 0, stream>>>(W, magsq, thresh, out);
}